// ESH_14491219657169
// MI455X (gfx1250) — compile-verified
//
#include <hip/hip_runtime.h>
#include <hip/hip_bf16.h>

// ESH integrator on gfx1250 (MI455X).
// 16 rows per wave32: lane n and lane n+16 hold the two 32-element halves of
// row n. Row reductions: per-lane FMA partials -> one V_WMMA_F32_16X16X4_F32
// with A = ones(16x4); the WMMA both sums across the lane pair and broadcasts
// the per-row result (exact f32, layout-permutation invariant since A==1).

typedef float v2f __attribute__((ext_vector_type(2)));
typedef float v4f __attribute__((ext_vector_type(4)));
typedef float v8f __attribute__((ext_vector_type(8)));

#define ESH_EPS_MIN 1e-10f
#define ESH_MAX_GN  10.0f

// Sum the four per-row partials (p0,p1 in lane n; p0,p1 in lane n+16) for each
// of the 16 rows and broadcast: returns row-sum of row (lane & 15) in every lane.
__device__ __forceinline__ float rowsum_wmma(float p0, float p1) {
    v2f a; a[0] = 1.0f; a[1] = 1.0f;        // A = ones(16x4)
    v2f b; b[0] = p0;   b[1] = p1;          // B = partials (4x16), N = lane&15
    v8f c = {};
    c = __builtin_amdgcn_wmma_f32_16x16x4_f32(
            /*neg_a=*/false, a, /*neg_b=*/false, b,
            /*c_mod=*/(short)0, c, /*reuse_a=*/false, /*reuse_b=*/false);
    return c[0];                            // D[m][lane&15], identical for all m
}

__device__ __forceinline__ void load32(const float* __restrict__ p, float (&a)[32]) {
    const v4f* v = reinterpret_cast<const v4f*>(p);
#pragma unroll
    for (int q = 0; q < 8; ++q) {
        v4f t = v[q];
        a[4*q+0] = t[0]; a[4*q+1] = t[1]; a[4*q+2] = t[2]; a[4*q+3] = t[3];
    }
}

__device__ __forceinline__ void store32_nt(float* __restrict__ p, const float (&a)[32]) {
    v4f* v = reinterpret_cast<v4f*>(p);
#pragma unroll
    for (int q = 0; q < 8; ++q) {
        v4f t;
        t[0] = a[4*q+0]; t[1] = a[4*q+1]; t[2] = a[4*q+2]; t[3] = a[4*q+3];
        __builtin_nontemporal_store(t, &v[q]);   // write-once trajectory: TH=NT
    }
}

// One u/r half-step. grad = pc*x recomputed on the fly (saves a 32-reg array).
__device__ __forceinline__ void half_step(const float (&x)[32], const float (&pc)[32],
                                          float (&u)[32], float& r, float e_over_d) {
    // reduction 1: ||grad||^2
    float p0 = 0.f, p1 = 0.f;
#pragma unroll
    for (int j = 0; j < 16; ++j) {
        float ga = pc[j]      * x[j];
        float gb = pc[j + 16] * x[j + 16];
        p0 = fmaf(ga, ga, p0);
        p1 = fmaf(gb, gb, p1);
    }
    float gsq    = rowsum_wmma(p0, p1);
    float gn0    = fmaxf(sqrtf(gsq), ESH_EPS_MIN);
    float g_norm = fminf(gn0, ESH_MAX_GN);
    float inv    = 1.0f / gn0;

    // reduction 2: sum(u * grad_e)  (u_dot_e = -sum)
    float q0 = 0.f, q1 = 0.f;
#pragma unroll
    for (int j = 0; j < 16; ++j) {
        q0 = fmaf(u[j],      (pc[j]      * x[j])      * inv, q0);
        q1 = fmaf(u[j + 16], (pc[j + 16] * x[j + 16]) * inv, q1);
    }
    float ud  = -rowsum_wmma(q0, q1);

    float egd = e_over_d * g_norm;          // e * g_norm / d
    float t   = expf(-egd);                 // exp(-e*g_norm/d)
    float A2  = (ud - 1.0f) * (t * t);      // (u_dot_e - 1) * exp(-2 e g/d)
    float A   = 1.0f + ud + A2;
    float Bc  = 2.0f * t;
    bool cond = ud > -0.999f;

    // u_new + reduction 3: ||u_new||^2
    float s0 = 0.f, s1 = 0.f;
#pragma unroll
    for (int j = 0; j < 32; ++j) {
        float gej  = (pc[j] * x[j]) * inv;
        float perp = fmaf(gej, ud, u[j]);              // u + grad_e * u_dot_e
        float un   = fmaf(Bc, perp, -(A * gej));       // A*(-ge) + Bc*perp
        un = cond ? un : gej;
        u[j] = un;
        if (j < 16) s0 = fmaf(un, un, s0); else s1 = fmaf(un, un, s1);
    }
    float nsq  = rowsum_wmma(s0, s1);
    float ninv = 1.0f / fmaxf(sqrtf(nsq), ESH_EPS_MIN);
#pragma unroll
    for (int j = 0; j < 32; ++j) u[j] *= ninv;

    float Z  = 1.0f + ud - A2;
    float dr = cond ? (egd + logf(0.5f * fmaxf(Z, ESH_EPS_MIN))) : -egd;
    r += dr;
}

__global__ void __launch_bounds__(256)
esh_run_kernel(const float* __restrict__ x0, const float* __restrict__ u0,
               const float* __restrict__ prec, const float* __restrict__ epsp,
               const int* __restrict__ nsp, float* __restrict__ out, int Btot) {
    const int lane  = threadIdx.x & 31;
    const int wave  = blockIdx.x * (blockDim.x >> 5) + (threadIdx.x >> 5);
    if (wave * 16 >= Btot) return;              // whole-wave exit: EXEC all-1s for WMMA

    const int row   = wave * 16 + (lane & 15);  // this lane's row
    const int kbase = (lane >> 4) * 32;         // which half of the row

    const float eps      = *epsp;
    const int   n_steps  = *nsp;
    const float e_over_d = (0.5f * eps) * (1.0f / 64.0f);   // e = eps/2, d = 64

    const size_t nrows = (size_t)(n_steps + 1) * (size_t)Btot;
    float* __restrict__ outX = out;
    float* __restrict__ outU = out + nrows * 64;
    float* __restrict__ outR = out + 2 * nrows * 64;

    float x[32], u[32], pc[32];
    load32(x0   + (size_t)row * 64 + kbase, x);
    load32(u0   + (size_t)row * 64 + kbase, u);
    load32(prec + kbase, pc);
    float r = 0.0f;

    // trajectory slot 0 = initial state
    store32_nt(outX + (size_t)row * 64 + kbase, x);
    store32_nt(outU + (size_t)row * 64 + kbase, u);
    if (lane < 16) outR[row] = 0.0f;

    for (int s = 0; s < n_steps; ++s) {
        half_step(x, pc, u, r, e_over_d);       // grad at current x
#pragma unroll
        for (int j = 0; j < 32; ++j) x[j] = fmaf(eps, u[j], x[j]);
        half_step(x, pc, u, r, e_over_d);       // grad at updated x

        const size_t roff = (size_t)(s + 1) * (size_t)Btot + (size_t)row;
        store32_nt(outX + roff * 64 + kbase, x);
        store32_nt(outU + roff * 64 + kbase, u);
        if (lane < 16) outR[roff] = r;
    }
}

extern "C" void kernel_launch(void* const* d_in, const int* in_sizes, int n_in,
                              void* d_out, int out_size, void* d_ws, size_t ws_size,
                              hipStream_t stream) {
    (void)n_in; (void)out_size; (void)d_ws; (void)ws_size;
    const float* x0   = (const float*)d_in[0];
    const float* u0   = (const float*)d_in[1];
    const float* prec = (const float*)d_in[2];
    const float* eps  = (const float*)d_in[3];
    const int*   ns   = (const int*)d_in[4];

    const int B       = in_sizes[0] / 64;       // rows (D = 64)
    const int waves   = (B + 15) / 16;          // 16 rows per wave32
    const int threads = waves * 32;
    const int block   = 256;                    // 8 waves / block
    const int grid    = (threads + block - 1) / block;

    esh_run_kernel<<<grid, block, 0, stream>>>(x0, u0, prec, eps, ns,
                                               (float*)d_out, B);
}